// DenseDilatedKnnGraph_42417097016503
// MI455X (gfx1250) — compile-verified
//
#include <hip/hip_runtime.h>
#include <hip/hip_bf16.h>

#define BATCH 8
#define CH 32
#define NPTS 4096
#define KALL 18
#define KOUT 9

typedef float v2f __attribute__((ext_vector_type(2)));
typedef float v4f __attribute__((ext_vector_type(4)));
typedef float v8f __attribute__((ext_vector_type(8)));

// ---------------------------------------------------------------------------
// Pre-pass: squared norms per point. norms[b*N + n] = sum_c x[b][c][n]^2
// x layout: (B, C, N, 1) row-major f32.
// ---------------------------------------------------------------------------
__global__ void __launch_bounds__(256) knn_norms_kernel(const float* __restrict__ x,
                                                        float* __restrict__ norms) {
    int t = blockIdx.x * blockDim.x + threadIdx.x;   // 0 .. B*N-1
    int b = t / NPTS;
    int n = t - b * NPTS;
    const float* xb = x + (size_t)b * CH * NPTS;
    float s = 0.0f;
#pragma unroll
    for (int c = 0; c < CH; ++c) {
        float v = xb[c * NPTS + n];
        s = fmaf(v, v, s);
    }
    norms[t] = s;
}

// ---------------------------------------------------------------------------
// Fused tile-GEMM (exact f32 WMMA) + streaming top-18 selection.
// One wave32 per (batch, 16-row tile). Never materializes the NxN matrix.
//
// The WMMA accumulator directly produces m = 2<xi,xj> - |xi|^2 - |xj|^2
// (= -dist): the 8 main K=4 steps use A pre-scaled by 2, plus one extra
// rank-2 K=4 step carrying (sqA)*(-1) and (-1)*(sqB). Selection then keeps
// the 18 LARGEST m per row, matching top_k(-dist).
// ---------------------------------------------------------------------------
__global__ void __launch_bounds__(32) knn_select_kernel(const float* __restrict__ x,
                                                        const float* __restrict__ norms,
                                                        int* __restrict__ out) {
    const int ntiles = NPTS / 16;                    // 256 column tiles
    int b  = blockIdx.x / ntiles;
    int it = blockIdx.x - b * ntiles;
    int i0 = it * 16;

    int L      = threadIdx.x;                        // 0..31 (wave32)
    int half   = L >> 4;                             // 0 or 1
    int lane16 = L & 15;                             // row this lane owns

    const float* xb = x + (size_t)b * CH * NPTS;
    const float* nb = norms + (size_t)b * NPTS;

    // A-matrix fragments: 16x4 f32 per step, 8 steps cover K=C=32.
    // Layout (ISA 7.12.2): lane L -> M = L%16; VGPR v -> K = v + 2*(L/16).
    // Pre-scaled by 2 so the accumulator carries 2*<xi,xj>.
    v2f afrag[8];
#pragma unroll
    for (int kk = 0; kk < 8; ++kk) {
        int k = 4 * kk + 2 * half;
        afrag[kk][0] = 2.0f * xb[(k + 0) * NPTS + i0 + lane16];
        afrag[kk][1] = 2.0f * xb[(k + 1) * NPTS + i0 + lane16];
    }
    float sqA = nb[i0 + lane16];

    // Extra rank-2 step: K=0 -> A=sqA[row], B=-1 ; K=1 -> A=-1, B=sqB[col].
    // Lanes 16..31 hold K=2,3 which are zero.
    v2f aextra;
    aextra[0] = (half == 0) ? sqA  : 0.0f;
    aextra[1] = (half == 0) ? -1.0f : 0.0f;

    // Register-resident sorted top-18 of m = -dist (descending).
    float bestd[KALL];
    int   besti[KALL];
#pragma unroll
    for (int s = 0; s < KALL; ++s) { bestd[s] = -3.0e38f; besti[s] = 0; }

    __shared__ __align__(16) float tile[16 * 16];   // staged -dist tile
    __shared__ float md[16 * KALL];                 // merge buffer (scores)
    __shared__ int   mi[16 * KALL];                 // merge buffer (indices)

    for (int jt = 0; jt < ntiles; ++jt) {
        int j0 = jt * 16;

        // B-matrix fragments (4x16 per step): symmetric layout, column = lane16.
        v2f bfrag[8];
#pragma unroll
        for (int kk = 0; kk < 8; ++kk) {
            int k = 4 * kk + 2 * half;
            bfrag[kk][0] = xb[(k + 0) * NPTS + j0 + lane16];
            bfrag[kk][1] = xb[(k + 1) * NPTS + j0 + lane16];
        }
        float sqB = nb[j0 + lane16];
        v2f bextra;
        bextra[0] = (half == 0) ? -1.0f : 0.0f;
        bextra[1] = (half == 0) ? sqB   : 0.0f;

        // 9 chained V_WMMA_F32_16X16X4_F32: acc = 2*inner - sqA - sqB = -dist.
        v8f acc = {};
        acc = __builtin_amdgcn_wmma_f32_16x16x4_f32(
            false, aextra, false, bextra, (short)0, acc, false, false);
#pragma unroll
        for (int kk = 0; kk < 8; ++kk) {
            acc = __builtin_amdgcn_wmma_f32_16x16x4_f32(
                false, afrag[kk], false, bfrag[kk], (short)0, acc, false, false);
        }

        // C/D layout: lane L, VGPR v -> row M = v + 8*(L/16), col N = L%16.
#pragma unroll
        for (int v = 0; v < 8; ++v) {
            tile[(v + 8 * half) * 16 + lane16] = acc[v];
        }
        __syncthreads();

        // Selection: lane r takes cols 0..7 of row r, lane r+16 cols 8..15.
        // 8 consecutive floats -> two b128 LDS loads.
        const v4f* tp = (const v4f*)&tile[lane16 * 16 + 8 * half];
        v4f t0 = tp[0];
        v4f t1 = tp[1];
        float tv[8];
#pragma unroll
        for (int c = 0; c < 4; ++c) { tv[c] = t0[c]; tv[c + 4] = t1[c]; }

#pragma unroll
        for (int c = 0; c < 8; ++c) {
            float m = tv[c];
            int   j = j0 + 8 * half + c;
            if (m > bestd[KALL - 1]) {
                bestd[KALL - 1] = m;
                besti[KALL - 1] = j;
#pragma unroll
                for (int s = KALL - 1; s > 0; --s) {
                    if (bestd[s] > bestd[s - 1]) {
                        float td = bestd[s]; bestd[s] = bestd[s - 1]; bestd[s - 1] = td;
                        int   ti = besti[s]; besti[s] = besti[s - 1]; besti[s - 1] = ti;
                    }
                }
            }
        }
        __syncthreads();   // protect tile reuse next iteration
    }

    // Merge the two half-lists per row.
    if (half == 1) {
#pragma unroll
        for (int s = 0; s < KALL; ++s) {
            md[lane16 * KALL + s] = bestd[s];
            mi[lane16 * KALL + s] = besti[s];
        }
    }
    __syncthreads();

    if (half == 0) {
#pragma unroll
        for (int s = 0; s < KALL; ++s) {
            float m = md[lane16 * KALL + s];
            int   j = mi[lane16 * KALL + s];
            if (m > bestd[KALL - 1]) {
                bestd[KALL - 1] = m;
                besti[KALL - 1] = j;
#pragma unroll
                for (int t = KALL - 1; t > 0; --t) {
                    if (bestd[t] > bestd[t - 1]) {
                        float td = bestd[t]; bestd[t] = bestd[t - 1]; bestd[t - 1] = td;
                        int   ti = besti[t]; besti[t] = besti[t - 1]; besti[t - 1] = ti;
                    }
                }
            }
        }

        // Emit dilated slice ::2 of the sorted 18 -> 9 neighbors.
        // Output shape (2, B, N, 9) int32, row-major.
        int    n     = i0 + lane16;
        size_t base0 = ((size_t)b * NPTS + n) * KOUT;
        size_t base1 = (size_t)BATCH * NPTS * KOUT + base0;
#pragma unroll
        for (int k = 0; k < KOUT; ++k) {
            out[base0 + k] = besti[2 * k];
            out[base1 + k] = n;
        }
    }
}

// ---------------------------------------------------------------------------
extern "C" void kernel_launch(void* const* d_in, const int* in_sizes, int n_in,
                              void* d_out, int out_size, void* d_ws, size_t ws_size,
                              hipStream_t stream) {
    const float* x     = (const float*)d_in[0];     // (8, 32, 4096, 1) f32
    float*       norms = (float*)d_ws;              // B*N floats = 128 KB scratch
    int*         out   = (int*)d_out;               // (2, 8, 4096, 9) int32

    // Pass 1: per-point squared norms.
    knn_norms_kernel<<<(BATCH * NPTS) / 256, 256, 0, stream>>>(x, norms);

    // Pass 2: fused WMMA (-dist) tiles + streaming top-18 + dilated emit.
    knn_select_kernel<<<BATCH * (NPTS / 16), 32, 0, stream>>>(x, norms, out);
}